// SparseChannelLinear_51290499449145
// MI455X (gfx1250) — compile-verified
//
#include <hip/hip_runtime.h>

typedef __attribute__((ext_vector_type(2))) float v2f;
typedef __attribute__((ext_vector_type(4))) float v4f;
typedef __attribute__((ext_vector_type(8))) float v8f;
typedef __attribute__((ext_vector_type(4))) int   v4i;

// Problem dims (match reference): N,C,H,O,K
#define NB 8
#define CB 4096
#define HB 768
#define OB 768
#define KB 1024

#define BK   32   // K-depth per LDS stage
#define LDST 36   // LDS row stride in floats (BK + 4 pad)

#if __has_builtin(__builtin_amdgcn_global_load_async_to_lds_b128) && \
    __has_builtin(__builtin_amdgcn_s_wait_asynccnt)
#define HAVE_ASYNC 1
#pragma message("CDNA5 async global->LDS path ENABLED")
#else
#define HAVE_ASYNC 0
#pragma message("CDNA5 async builtins NOT available; using sync staging fallback")
#endif

typedef __attribute__((address_space(1))) v4i* gptr_b128;  // global (AS1)
typedef __attribute__((address_space(3))) v4i* lptr_b128;  // LDS    (AS3)

__device__ __forceinline__ void stage16(const float* __restrict__ g, float* l) {
#if HAVE_ASYNC
    __builtin_amdgcn_global_load_async_to_lds_b128(
        (gptr_b128)(void*)const_cast<float*>(g),
        (lptr_b128)(void*)l,
        /*offset=*/0, /*cpol=*/0);
#else
    *(float4*)l = *(const float4*)g;
#endif
}

__device__ __forceinline__ void stage_wait() {
#if HAVE_ASYNC
    __builtin_amdgcn_s_wait_asynccnt(0);
#endif
}

__global__ void scl_zero(v4f* __restrict__ out, long n4) {
    long i = (long)blockIdx.x * blockDim.x + threadIdx.x;
    long stride = (long)gridDim.x * blockDim.x;
    v4f z = {};
    for (; i < n4; i += stride)
        __builtin_nontemporal_store(z, &out[i]);
}

__global__ __launch_bounds__(256) void scl_gemm(
    const float* __restrict__ input,   // [N, C, H]
    const float* __restrict__ weight,  // [O, H]
    const float* __restrict__ bias,    // [O]
    const int*   __restrict__ cidx,    // [N, K, 1]
    float*       __restrict__ out)     // [N, C, O]
{
    __shared__ float ldsA[2][128 * LDST];
    __shared__ float ldsW[2][128 * LDST];
    __shared__ int   sidx[128];

    const int tid   = threadIdx.x;
    const int lane  = tid & 31;
    const int wave  = tid >> 5;
    const int oTile = blockIdx.x;   // 0..5   -> output columns [oTile*128, +128)
    const int kTile = blockIdx.y;   // 0..7   -> gathered rows  [kTile*128, +128)
    const int batch = blockIdx.z;   // 0..7

    // 8 waves arranged 4 (M) x 2 (N); each wave computes 32x64
    const int mBase  = (wave & 3) * 32;
    const int nBase  = (wave >> 2) * 64;
    const int lane15 = lane & 15;
    const int klo    = (lane >> 4) << 1;   // lanes 16..31 hold K+2, K+3 (ISA 7.12.2)

    // stage channel indices for this M tile
    if (tid < 128) sidx[tid] = cidx[batch * KB + kTile * 128 + tid];
    __syncthreads();

    // per-thread global->LDS staging coords: 256 thr x 4 rows x float4 = 128x32
    const int lr  = tid >> 3;        // row group 0..31
    const int lc4 = (tid & 7) * 4;   // column (float) within BK

    const float* wBase = weight + (size_t)(oTile * 128) * HB;
    const float* xBase = input  + (size_t)batch * CB * HB;

    // loop-invariant gather/weight row base pointers + LDS destinations
    const float* aSrc[4];
    const float* wSrc[4];
    int ldsOff[4];
    #pragma unroll
    for (int i = 0; i < 4; ++i) {
        const int row = lr + 32 * i;
        aSrc[i]   = xBase + (size_t)sidx[row] * HB + lc4;
        wSrc[i]   = wBase + (size_t)row * HB + lc4;
        ldsOff[i] = row * LDST + lc4;
    }

    auto stage_block = [&](int buf, int k0) {
        #pragma unroll
        for (int i = 0; i < 4; ++i) {
            stage16(aSrc[i] + k0, &ldsA[buf][ldsOff[i]]);
            stage16(wSrc[i] + k0, &ldsW[buf][ldsOff[i]]);
        }
    };

    v8f acc[2][4] = {};

    stage_block(0, 0);
    stage_wait();
    __syncthreads();

    constexpr int NITER = HB / BK;   // 24
    for (int it = 0; it < NITER; ++it) {
        const int cur = it & 1;
        if (it + 1 < NITER)
            stage_block(cur ^ 1, (it + 1) * BK);   // async prefetch overlaps WMMAs

        const float* lA = &ldsA[cur][0];
        const float* lW = &ldsW[cur][0];
        #pragma unroll
        for (int kk = 0; kk < BK; kk += 4) {
            v2f afrag[2], bfrag[4];
            #pragma unroll
            for (int mi = 0; mi < 2; ++mi)
                afrag[mi] = *(const v2f*)(&lA[(mBase + mi * 16 + lane15) * LDST + kk + klo]);
            #pragma unroll
            for (int ni = 0; ni < 4; ++ni)
                bfrag[ni] = *(const v2f*)(&lW[(nBase + ni * 16 + lane15) * LDST + kk + klo]);
            #pragma unroll
            for (int mi = 0; mi < 2; ++mi)
                #pragma unroll
                for (int ni = 0; ni < 4; ++ni)
                    acc[mi][ni] = __builtin_amdgcn_wmma_f32_16x16x4_f32(
                        false, afrag[mi], false, bfrag[ni],
                        (short)0, acc[mi][ni], false, false);
        }

        stage_wait();      // my prefetch into buf^1 has landed in LDS
        __syncthreads();   // all waves done computing buf + staging buf^1
    }

    // Epilogue: bias add + scatter by gathered row index (write-once -> non-temporal).
    // C/D layout: VGPR r -> lanes 0-15: M=r, lanes 16-31: M=8+r; N = lane&15.
    const int rhalf = (lane >> 4) * 8;
    float bvals[4];
    #pragma unroll
    for (int ni = 0; ni < 4; ++ni)
        bvals[ni] = bias[oTile * 128 + nBase + ni * 16 + lane15];

    #pragma unroll
    for (int mi = 0; mi < 2; ++mi) {
        #pragma unroll
        for (int r = 0; r < 8; ++r) {
            const int rowL = mBase + mi * 16 + rhalf + r;
            const int gRow = sidx[rowL];
            float* orow = out + ((size_t)batch * CB + gRow) * OB + oTile * 128;
            #pragma unroll
            for (int ni = 0; ni < 4; ++ni)
                __builtin_nontemporal_store(acc[mi][ni][r] + bvals[ni],
                                            &orow[nBase + ni * 16 + lane15]);
        }
    }
}

extern "C" void kernel_launch(void* const* d_in, const int* in_sizes, int n_in,
                              void* d_out, int out_size, void* d_ws, size_t ws_size,
                              hipStream_t stream) {
    const float* input  = (const float*)d_in[0];
    const float* weight = (const float*)d_in[1];
    const float* bias   = (const float*)d_in[2];
    const int*   cidx   = (const int*)d_in[3];
    float*       out    = (float*)d_out;

    // 1) zero-init the full [N, C, O] output (scatter target)
    long n4 = (long)out_size / 4;   // 25165824 / 4
    scl_zero<<<dim3(2048), dim3(256), 0, stream>>>((v4f*)out, n4);

    // 2) gathered GEMM + scatter (stream-ordered after zero-fill)
    dim3 grid(OB / 128, KB / 128, NB);   // (6, 8, 8)
    scl_gemm<<<grid, dim3(256), 0, stream>>>(input, weight, bias, cidx, out);
}